// NeedlemanWunschDecoder_32770600468778
// MI455X (gfx1250) — compile-verified
//
#include <hip/hip_runtime.h>
#include <stdint.h>

// Needleman-Wunsch forward DP (logsumexp operator), B=32, N=M=512.
// Anti-diagonal wavefront; 1 workgroup/batch; 256 threads x 2 columns.
// Waves are time-skewed by 16 diagonals so the workgroup barrier is needed
// only once per 16 steps (epoch). DP state lives in registers; intra-wave
// neighbor via shuffle; cross-wave seam via a 32-deep LDS ring; theta is
// register-pipelined 2 diagonals deep; TDM streams theta through L2.

#define NEGV (-1e10f)
#define NN 512
#define MM 512
#define NT 256            // threads per block
#define NWAVE (NT / 32)   // 8 waves
#define SKEW 16           // diagonal lag between adjacent waves
#define EPOCH 16          // steps between workgroup barriers
#define RINGD 32          // seam ring depth (2 * EPOCH)
#define TOTAL (NN + MM - 1 + SKEW * (NWAVE - 1) + 1)   // 1136 steps

typedef unsigned int v4u __attribute__((ext_vector_type(4)));
typedef int          v4i __attribute__((ext_vector_type(4)));
typedef int          v8i __attribute__((ext_vector_type(8)));

__device__ __forceinline__ float lse3(float a, float b, float c) {
    float m = fmaxf(a, fmaxf(b, c));
    float s = __expf(a - m) + __expf(b - m) + __expf(c - m); // v_exp_f32
    return m + __logf(s);                                    // v_log_f32
}

__global__ __launch_bounds__(NT, 1) void nw_diag_kernel(
        const float* __restrict__ theta,
        const float* __restrict__ A,
        float* __restrict__ out)
{
    const int b    = blockIdx.x;
    const int tid  = threadIdx.x;
    const int lane = tid & 31;
    const int w    = tid >> 5;
    const int j0   = 2 * tid + 1;   // first owned DP column
    const int j1   = 2 * tid + 2;   // second owned DP column
    const float gap = A[b];
    const float* __restrict__ th = theta + (size_t)b * NN * MM;

    // Cross-wave seam ring: ring[w][d & 31] = V[col 64w+64, diag d].
    __shared__ float ring[NWAVE][RINGD];
    // TDM staging ring (write-only; DMA stream warms L2 for strided gathers).
    __shared__ float stage[2][8 * MM];

    ((float*)ring)[tid] = NEGV;      // NT == NWAVE*RINGD == 256

    // ---- Tensor Data Mover: stream this batch's theta (1 MB) through L2 ----
    // Issued by the most-delayed wave (idle for its first 112 skew steps).
#if __has_builtin(__builtin_amdgcn_tensor_load_to_lds)
    if (tid >= NT - 32) {
        const uint64_t gbase = (uint64_t)(uintptr_t)th;
        const uint32_t lds0  = (uint32_t)(uintptr_t)&stage[0][0];
        const uint32_t lds1  = (uint32_t)(uintptr_t)&stage[1][0];
        for (int p = 0; p < NN / 8; ++p) {               // 64 tiles of 8 rows
            uint64_t ga = gbase + (uint64_t)p * 8 * MM * 4;
            v4u g0 = {0u, 0u, 0u, 0u};
            g0[0] = 1u;                                  // count=1 (valid D#)
            g0[1] = (p & 1) ? lds1 : lds0;               // lds_addr (bytes)
            g0[2] = (uint32_t)ga;                        // global_addr[31:0]
            g0[3] = (uint32_t)(ga >> 32) | (2u << 30);   // addr[56:32] | type=2
            v8i g1 = {0, 0, 0, 0, 0, 0, 0, 0};
            g1[0] = 0x00020000;                          // data_size=4B
            g1[1] = (int)((MM & 0xFFFF) << 16);          // tensor_dim0 lo16
            g1[2] = (int)((MM >> 16) | (8u << 16));      // dim0 hi | tensor_dim1=8
            g1[3] = (int)((MM & 0xFFFF) << 16);          // dim1 hi | tile_dim0=512
            g1[4] = 8;                                   // tile_dim1=8
            g1[5] = MM;                                  // tensor_dim0_stride
            v4i g2 = {0, 0, 0, 0};
            v4i g3 = {0, 0, 0, 0};
#if __clang_major__ >= 23
            __builtin_amdgcn_tensor_load_to_lds(g0, g1, g2, g3, (v8i){0,0,0,0,0,0,0,0}, 0);
#else
            __builtin_amdgcn_tensor_load_to_lds(g0, g1, g2, g3, 0);
#endif
        }
    }
#endif

    // Register-resident DP state for the two owned columns.
    // v1a/v1b: cols j0/j1 on diagonal d-1; v2a/v2b: diagonal d-2.
    float v1a = NEGV, v1b = NEGV, v2a = NEGV, v2b = NEGV;
    // Seam column j0-1 at diagonal d-2 (carried register). For thread 0 at
    // d=2 this is cell (0,0)=0.
    float nv2 = (tid == 0) ? 0.0f : NEGV;
    // Lane-0 seam prefetch register: ring value for seam diagonal d-1,
    // loaded one step early (LDS latency never feeds the LSE chain).
    float nvr = NEGV;

    // Theta pipeline: *_c -> diagonal d, *_n -> d+1.
    float t0c = 0.0f, t1c = 0.0f, t0n = 0.0f, t1n = 0.0f;
    {
        int dp = 2 - SKEW * w;       // diagonal consumed at step 0
        int i0 = dp - j0;     if (i0 >= 1 && i0 <= NN) t0c = th[(size_t)(i0 - 1) * MM + (j0 - 1)];
        int i1 = dp - j1;     if (i1 >= 1 && i1 <= NN) t1c = th[(size_t)(i1 - 1) * MM + (j1 - 1)];
        int k0 = dp + 1 - j0; if (k0 >= 1 && k0 <= NN) t0n = th[(size_t)(k0 - 1) * MM + (j0 - 1)];
        int k1 = dp + 1 - j1; if (k1 >= 1 && k1 <= NN) t1n = th[(size_t)(k1 - 1) * MM + (j1 - 1)];
    }

    __syncthreads();

    for (int step = 0; step < TOTAL; ++step) {
        const int d = step + 2 - SKEW * w;   // this wave's diagonal

        // Issue theta loads for diagonal d+2 (hidden behind ~2 full steps).
        float t0f = 0.0f, t1f = 0.0f;
        {
            int i0 = d + 2 - j0; if (i0 >= 1 && i0 <= NN) t0f = th[(size_t)(i0 - 1) * MM + (j0 - 1)];
            int i1 = d + 2 - j1; if (i1 >= 1 && i1 <= NN) t1f = th[(size_t)(i1 - 1) * MM + (j1 - 1)];
        }
        // gfx1250 global_prefetch_b8: warm WGP$ 8 diagonals ahead.
        {
            int i0 = d + 8 - j0;
            if (i0 >= 1 && i0 <= NN)
                __builtin_prefetch(&th[(size_t)(i0 - 1) * MM + (j0 - 1)], 0, 1);
        }

        // Seam column j0-1 at diagonal d-1: one lane away inside the wave,
        // or the early-prefetched ring value for lane 0.
        float nv1 = __shfl_up(v1b, 1);
        if (lane == 0)
            nv1 = (w > 0) ? nvr : NEGV;      // column-0 boundary for wave 0

        const int i0 = d - j0, i1 = d - j1;
        // Cell (i0,j0): up=(i0-1,j0)=v1a, diag=(i0-1,j0-1)=nv2, left=(i0,j0-1)=nv1
        float x0 = (i0 >= 1 && i0 <= NN)
                     ? (t0c + lse3(gap + v1a, nv2, gap + nv1)) : NEGV;
        // Cell (i1,j1): up=(i1-1,j1)=v1b, diag=(i1-1,j0)=v2a, left=(i1,j0)=v1a
        float x1 = (i1 >= 1 && i1 <= NN)
                     ? (t1c + lse3(gap + v1b, v2a, gap + v1a)) : NEGV;

        // Rotate register state to diagonal d.
        v2a = v1a; v2b = v1b;
        v1a = x0;  v1b = x1;
        nv2 = nv1;                           // seam d-1 becomes next step's d-2

        // Publish seam value V[col 64w+64, diag d] (written 16+ steps before
        // its consumer reads it; slot reused 32 diagonals later).
        if (lane == 31 && d >= 2 && d <= NN + MM)
            ring[w][d & (RINGD - 1)] = v1b;

        // Early seam load for NEXT step (seam diagonal = current d); the
        // producer wrote it 16 steps ago, i.e. at least one epoch barrier ago.
        if (lane == 0 && w > 0 && d >= 2)
            nvr = ring[w - 1][d & (RINGD - 1)];

        // Epoch barrier: orders seam ring writes/reads across waves.
        if ((step & (EPOCH - 1)) == (EPOCH - 1))
            __syncthreads();

        t0c = t0n; t1c = t1n; t0n = t0f; t1n = t1f;
    }

    // Wave 7 finishes last: its final diagonal is N+M+1, so v1b here is NEGV (cell (N, M) sits in v2b).
    if (tid == NT - 1) out[b] = v1b;

#if __has_builtin(__builtin_amdgcn_s_wait_tensorcnt)
    if (tid >= NT - 32) __builtin_amdgcn_s_wait_tensorcnt(0);
#endif
}

extern "C" void kernel_launch(void* const* d_in, const int* in_sizes, int n_in,
                              void* d_out, int out_size, void* d_ws, size_t ws_size,
                              hipStream_t stream) {
    (void)n_in; (void)d_ws; (void)ws_size; (void)out_size;
    const float* theta = (const float*)d_in[0];
    const float* A     = (const float*)d_in[1];
    float*       out   = (float*)d_out;
    const int B = in_sizes[1];   // 32 batches
    nw_diag_kernel<<<dim3(B), dim3(NT), 0, stream>>>(theta, A, out);
}